// SO3TensorProduct_29368986370622
// MI455X (gfx1250) — compile-verified
//
#include <hip/hip_runtime.h>
#include <hip/hip_bf16.h>

typedef __attribute__((ext_vector_type(2))) float v2f;
typedef __attribute__((ext_vector_type(8))) float v8f;

#define S9   9
#define NFEAT 128
#define KPAIR 81          // 9*9 dense (s1,s2) pairs
#define KPAD  84          // padded to multiple of 4 for 16x16x4 WMMA
#define NCHUNK (KPAD / 4) // 21 WMMA steps

// One block per atom; 8 waves per block, one 16-feature tile per wave.
// out[a, o, f] = sum_{s1,s2} C[o,s1,s2] * x1[a,s1,f] * x2[a,s2,f]
// computed as a 16x16x84 f32 WMMA matmul: A = C (LDS), B = pair products (regs).
__global__ __launch_bounds__(256)
void so3_tp_wmma_kernel(const float* __restrict__ x1,
                        const float* __restrict__ x2,
                        const float* __restrict__ cg,
                        const int*   __restrict__ i1,
                        const int*   __restrict__ i2,
                        const int*   __restrict__ io,
                        float* __restrict__ out,
                        int nnz)
{
    // A-matrix in LDS, layout A[k][o], k = s1*9+s2 in 0..83 (zero padded), o in 0..15.
    __shared__ float Alds[KPAD * 16];

    const int tid = threadIdx.x;

    // Build dense CG operator once per block (triples are unique -> no collisions).
    for (int idx = tid; idx < KPAD * 16; idx += 256) Alds[idx] = 0.0f;
    __syncthreads();
    for (int t = tid; t < nnz; t += 256) {
        const int k = i1[t] * S9 + i2[t];
        Alds[k * 16 + io[t]] = cg[t];
    }
    __syncthreads();

    const int wave = tid >> 5;        // feature tile 0..7
    const int lane = tid & 31;
    const int n    = lane & 15;       // column / A-row within half-wave
    const int h    = lane >> 4;       // lane half (selects K pair {0,1} vs {2,3})
    const bool hi  = (h != 0);

    const size_t a  = blockIdx.x;
    const int    f0 = wave * 16;

    const float* p1 = x1 + (a * S9) * NFEAT + f0 + n;
    const float* p2 = x2 + (a * S9) * NFEAT + f0 + n;

    // Stage the 9+9 inputs for this (atom, feature column) in registers.
    float X1[S9], X2[S9];
#pragma unroll
    for (int s = 0; s < S9; ++s) {
        X1[s] = __builtin_nontemporal_load(p1 + s * NFEAT);
        X2[s] = __builtin_nontemporal_load(p2 + s * NFEAT);
    }

    v8f acc = {0.f, 0.f, 0.f, 0.f, 0.f, 0.f, 0.f, 0.f};

    const int hoff = 32 * h;          // A LDS offset for lane half: (k0+2h)*16
#pragma unroll
    for (int kc = 0; kc < NCHUNK; ++kc) {
        const int k0 = kc * 4;

        // Pair products for the 4 K-values of this chunk (compile-time indices).
        float p[4];
#pragma unroll
        for (int j = 0; j < 4; ++j) {
            const int k = k0 + j;
            p[j] = (k < KPAIR) ? (X1[k / S9] * X2[k % S9]) : 0.0f;
        }
        // B 4x16 layout: VGPR0 = row k0 (lanes 0-15) / k0+2 (lanes 16-31), VGPR1 = k0+1 / k0+3.
        v2f B;
        B.x = hi ? p[2] : p[0];
        B.y = hi ? p[3] : p[1];

        // A 16x4 layout: VGPR0 = K k0 (lanes 0-15) / k0+2 (16-31), VGPR1 = k0+1 / k0+3.
        v2f A;
        A.x = Alds[(k0    ) * 16 + hoff + n];
        A.y = Alds[(k0 + 1) * 16 + hoff + n];

        acc = __builtin_amdgcn_wmma_f32_16x16x4_f32(
            /*neg_a=*/false, A, /*neg_b=*/false, B,
            /*c_mod=*/(short)0, acc, /*reuse_a=*/false, /*reuse_b=*/false);
    }

    // C/D layout: VGPR j -> row j (lanes 0-15), row j+8 (lanes 16-31). Rows 0..8 are valid.
    float* po = out + (a * S9) * NFEAT + f0 + n;
#pragma unroll
    for (int j = 0; j < 8; ++j) {
        const int m = j + 8 * h;
        if (m < S9) __builtin_nontemporal_store(acc[j], po + m * NFEAT);
    }
}

extern "C" void kernel_launch(void* const* d_in, const int* in_sizes, int n_in,
                              void* d_out, int out_size, void* d_ws, size_t ws_size,
                              hipStream_t stream)
{
    const float* x1 = (const float*)d_in[0];
    const float* x2 = (const float*)d_in[1];
    const float* cg = (const float*)d_in[2];
    const int*   i1 = (const int*)d_in[3];
    const int*   i2 = (const int*)d_in[4];
    const int*   io = (const int*)d_in[5];
    float*       out = (float*)d_out;

    const int nnz     = in_sizes[2];
    const int n_atoms = in_sizes[0] / (S9 * NFEAT);

    so3_tp_wmma_kernel<<<n_atoms, 256, 0, stream>>>(x1, x2, cg, i1, i2, io, out, nnz);
}